// Embedding_Pooled_Classifier_33303176413882
// MI455X (gfx1250) — compile-verified
//
#include <hip/hip_runtime.h>
#include <hip/hip_bf16.h>
#include <math.h>

// Problem constants (static per reference setup_inputs)
#define N_SUBDOCS 4096
#define N_WORDS   128
#define DIM       768
#define N_DOCS    256
#define N_ROWS    (N_SUBDOCS * N_WORDS)   // 524288 dot products of length DIM

typedef __attribute__((ext_vector_type(2))) float v2f;
typedef __attribute__((ext_vector_type(8))) float v8f;

// ---------------------------------------------------------------------------
// Kernel 1: z[row] = dot(E[row,:], W)  via V_WMMA_F32_16X16X4_F32 (exact fp32)
//
// A (16x4, SRC0)  : 16 consecutive rows of E, K-chunk of 4.
//   ISA layout: lane m (0-15): V0=K0,V1=K1 ; lane m+16: V0=K2,V1=K3.
//   => lane l loads E[row(l&15)][4c + 2*(l>>4) + {0,1}]  (8B contiguous).
// B (4x16, SRC1)  : every column n = W chunk -> D[m][n] identical over n.
//   transpose-symmetric layout => lane l supplies W[4c + 2*(l>>4) + {0,1}].
// D (16x16 f32)   : lane l, VGPR j -> row j + 8*(l>>4); lanes l%16==0 store
//   the 8 dots (all 16 columns hold the same value).
// ---------------------------------------------------------------------------
__global__ __launch_bounds__(256) void
epc_gemv_wmma(const float* __restrict__ emb,
              const float* __restrict__ W,
              float* __restrict__ z)
{
    __shared__ __align__(16) float lds_w[DIM];
    for (int i = threadIdx.x; i < DIM; i += 256) lds_w[i] = W[i];
    __syncthreads();

    const int lane = threadIdx.x & 31;
    const int wid  = threadIdx.x >> 5;
    const int tile = blockIdx.x * 8 + wid;        // 16 rows per tile, 8 waves/block
    const int m    = lane & 15;
    const int h    = lane >> 4;

    const float* rowp = emb + (size_t)(tile * 16 + m) * DIM + 2 * h;
    const float* wp   = lds_w + 2 * h;

    v8f acc = {0.f, 0.f, 0.f, 0.f, 0.f, 0.f, 0.f, 0.f};

#pragma unroll 4
    for (int k = 0; k < DIM; k += 4) {
        float2 av = *(const float2*)(rowp + k);   // global_load_b64, coalesced per-row
        float2 bv = *(const float2*)(wp + k);     // ds_load_b64 from LDS
        v2f a, b;
        a.x = av.x; a.y = av.y;
        b.x = bv.x; b.y = bv.y;
        // 8 args: (neg_a, A, neg_b, B, c_mod, C, reuse_a, reuse_b)
        acc = __builtin_amdgcn_wmma_f32_16x16x4_f32(
            false, a, false, b, (short)0, acc, false, false);
    }

    if (m == 0) {
        float* o = z + (size_t)tile * 16 + 8 * h; // 32B-aligned
        o[0] = acc[0]; o[1] = acc[1]; o[2] = acc[2]; o[3] = acc[3];
        o[4] = acc[4]; o[5] = acc[5]; o[6] = acc[6]; o[7] = acc[7];
    }
}

// ---------------------------------------------------------------------------
// Kernel 2: softmax-pool over the 128 words of each subdoc (one wave32 each).
//   x = z + bias ; zp[s] = sum(x * softmax(t*x))
// ---------------------------------------------------------------------------
__global__ __launch_bounds__(256) void
epc_pool_words(const float* __restrict__ z,
               const float* __restrict__ bptr,
               const float* __restrict__ logt,
               float* __restrict__ zp)
{
    const float t    = expf(logt[0]);
    const float bias = bptr[0];
    const int lane   = threadIdx.x & 31;
    const int s      = blockIdx.x * 8 + (threadIdx.x >> 5);

    const float* row = z + (size_t)s * N_WORDS;
    float x[4];
    float mx = -INFINITY;
#pragma unroll
    for (int j = 0; j < 4; ++j) {
        x[j] = row[lane + 32 * j] + bias;
        mx = fmaxf(mx, t * x[j]);
    }
    for (int off = 16; off; off >>= 1) mx = fmaxf(mx, __shfl_xor(mx, off, 32));

    float se = 0.f, sw = 0.f;
#pragma unroll
    for (int j = 0; j < 4; ++j) {
        float e = expf(t * x[j] - mx);
        se += e;
        sw += x[j] * e;
    }
    for (int off = 16; off; off >>= 1) {
        se += __shfl_xor(se, off, 32);
        sw += __shfl_xor(sw, off, 32);
    }
    if (lane == 0) zp[s] = sw / se;
}

// ---------------------------------------------------------------------------
// Kernel 3: ragged segment softmax-pool over subdocs + sigmoid (256 docs).
// One thread per doc; honors subdoc_lengths via exclusive prefix sum.
// ---------------------------------------------------------------------------
__global__ __launch_bounds__(256) void
epc_pool_docs(const float* __restrict__ zp,
              const int* __restrict__ lens,
              const float* __restrict__ logt,
              float* __restrict__ out)
{
    const int d = blockIdx.x * blockDim.x + threadIdx.x;
    if (d >= N_DOCS) return;

    const float t = expf(logt[0]);
    int off = 0;
    for (int j = 0; j < d; ++j) off += lens[j];
    const int len = lens[d];

    float mx = -INFINITY;
    for (int i = 0; i < len; ++i) mx = fmaxf(mx, t * zp[off + i]);
    float se = 0.f, sw = 0.f;
    for (int i = 0; i < len; ++i) {
        float v = zp[off + i];
        float e = expf(t * v - mx);
        se += e;
        sw += v * e;
    }
    float r = sw / se;
    out[d] = 1.f / (1.f + expf(-r));
}

// ---------------------------------------------------------------------------
extern "C" void kernel_launch(void* const* d_in, const int* in_sizes, int n_in,
                              void* d_out, int out_size, void* d_ws, size_t ws_size,
                              hipStream_t stream) {
    (void)in_sizes; (void)n_in; (void)out_size; (void)ws_size;

    const float* emb  = (const float*)d_in[0];   // [4096,128,768] f32
    const float* W    = (const float*)d_in[1];   // [1,768] f32
    const float* b    = (const float*)d_in[2];   // [1] f32
    const float* logt = (const float*)d_in[3];   // [1] f32
    const int*   lens = (const int*)d_in[4];     // [256] i32
    float*       out  = (float*)d_out;           // [256] f32

    float* z  = (float*)d_ws;                    // 524288 floats (2 MB)
    float* zp = z + N_ROWS;                      // 4096 floats

    // 32768 tiles of 16 rows; 8 waves (256 threads) per block
    epc_gemv_wmma<<<N_ROWS / 16 / 8, 256, 0, stream>>>(emb, W, z);
    // 4096 subdocs; one wave each, 8 per block
    epc_pool_words<<<N_SUBDOCS / 8, 256, 0, stream>>>(z, b, logt, zp);
    // 256 docs; one thread each
    epc_pool_docs<<<1, 256, 0, stream>>>(zp, lens, logt, out);
}